// GNNModel_33663953666137
// MI455X (gfx1250) — compile-verified
//
#include <hip/hip_runtime.h>
#include <hip/hip_bf16.h>

// ---------------------------------------------------------------------------
// GCN forward for MI455X (gfx1250, wave32).
// Dense GEMMs: V_WMMA_F32_16X16X4_F32 (full-precision f32 matrix path).
// A-tiles staged to LDS via GLOBAL_LOAD_ASYNC_TO_LDS_B128 (ASYNCcnt path).
// Sparse propagation: wave-per-edge float4 gathers + global f32 atomics
// (h matrix = 51 MB, resident in the 192 MB L2).
// ---------------------------------------------------------------------------

typedef float v2f __attribute__((ext_vector_type(2)));
typedef float v8f __attribute__((ext_vector_type(8)));
typedef int   v4i __attribute__((ext_vector_type(4)));

#define HDIM 128
#define FIN  256

#if defined(__HIP_DEVICE_COMPILE__) && __has_builtin(__builtin_amdgcn_global_load_async_to_lds_b128)
#define USE_ASYNC_LDS 1
typedef v4i __attribute__((address_space(1)))* g_v4i_p;  // global int4*
typedef v4i __attribute__((address_space(3)))* l_v4i_p;  // LDS int4*
#else
#define USE_ASYNC_LDS 0
#endif

__device__ __forceinline__ void wait_async0() {
#if __has_builtin(__builtin_amdgcn_s_wait_asynccnt)
    __builtin_amdgcn_s_wait_asynccnt(0);
#else
    asm volatile("s_wait_asynccnt 0x0" ::: "memory");
#endif
}

__device__ __forceinline__ float selu_f(float x) {
    const float alpha = 1.6732632423543772f;
    const float scale = 1.0507009873554805f;
    return x > 0.0f ? scale * x : scale * alpha * (__expf(x) - 1.0f);
}

// Stage a 16 x K f32 tile (rows m0..m0+15, clamped to Nn) into LDS with
// leading dimension K+4 (keeps rows 16B-aligned and A-reads bank-conflict
// free: bank = 4*lane + k mod 64, distinct across all 32 lanes).
template <int K>
__device__ __forceinline__ void stage_tile(const float* __restrict__ X,
                                           float* As, int m0, int Nn) {
    constexpr int LDA = K + 4;
#if USE_ASYNC_LDS
    constexpr int CHUNKS = 16 * (K / 4);  // one b128 (4 floats) per chunk
    for (int ch = threadIdx.x; ch < CHUNKS; ch += 256) {
        const int r  = ch / (K / 4);
        const int c4 = (ch % (K / 4)) * 4;
        const int row = min(m0 + r, Nn - 1);
        const float* gsrc = X + (size_t)row * K + c4;
        float*       ldst = As + r * LDA + c4;
        __builtin_amdgcn_global_load_async_to_lds_b128(
            (g_v4i_p)(uintptr_t)gsrc, (l_v4i_p)(uintptr_t)ldst, 0, 0);
    }
    wait_async0();
#else
    for (int idx = threadIdx.x; idx < 16 * K; idx += 256) {
        const int r = idx / K, c = idx % K;
        const int row = min(m0 + r, Nn - 1);
        As[r * LDA + c] = X[(size_t)row * K + c];
    }
#endif
    __syncthreads();
}

// Wave computes one 16x16 f32 tile: rows from LDS tile As (leading dim K+4),
// cols [n0, n0+16) of row-major W (K x ncol). K must be a multiple of 4.
// A layout (ISA 7.12.2, 32-bit A 16x4): lanes 0-15 hold K=k0,k0+1; lanes 16-31
// hold K=k0+2,k0+3 for row M=lane%16.  B mirrors with N striped across lanes.
template <int K>
__device__ __forceinline__ v8f wmma_tile_f32(const float* __restrict__ As,
                                             const float* __restrict__ W,
                                             int ncol, int n0, int lane) {
    constexpr int LDA = K + 4;
    const int half = lane >> 4;
    const int mr   = lane & 15;
    v8f acc = {0.f, 0.f, 0.f, 0.f, 0.f, 0.f, 0.f, 0.f};
#pragma unroll 8
    for (int k0 = 0; k0 < K; k0 += 4) {
        const int ka = k0 + 2 * half;
        v2f a, b;
        a.x = As[mr * LDA + ka];
        a.y = As[mr * LDA + ka + 1];
        b.x = W[(size_t)ka * ncol + n0 + mr];
        b.y = W[(size_t)(ka + 1) * ncol + n0 + mr];
        acc = __builtin_amdgcn_wmma_f32_16x16x4_f32(
            /*neg_a=*/false, a, /*neg_b=*/false, b,
            /*c_mod=*/(short)0, acc, /*reuse_a=*/false, /*reuse_b=*/false);
    }
    return acc;
}

// ---- Layer 0: h0 = selu(x @ W_red + b_red), K=256, ncol=128 ----------------
__global__ void k_gemm_selu(const float* __restrict__ X,
                            const float* __restrict__ W,
                            const float* __restrict__ bias,
                            float* __restrict__ out, int Nn) {
    __shared__ float As[16 * (FIN + 4)];
    const int m0 = blockIdx.x * 16;
    stage_tile<FIN>(X, As, m0, Nn);
    const int wave = threadIdx.x >> 5, lane = threadIdx.x & 31;
    const int n0 = wave * 16;
    v8f acc = wmma_tile_f32<FIN>(As, W, HDIM, n0, lane);
    const int half = lane >> 4, mr = lane & 15;
    const float bv = bias[n0 + mr];
#pragma unroll
    for (int r = 0; r < 8; ++r) {
        const int m = m0 + r + 8 * half;
        if (m < Nn) out[(size_t)m * HDIM + n0 + mr] = selu_f(acc[r] + bv);
    }
}

// ---- Mid GEMM: t = h @ W, K=128, ncol=128, no bias -------------------------
__global__ void k_gemm_h(const float* __restrict__ X,
                         const float* __restrict__ W,
                         float* __restrict__ out, int Nn) {
    __shared__ float As[16 * (HDIM + 4)];
    const int m0 = blockIdx.x * 16;
    stage_tile<HDIM>(X, As, m0, Nn);
    const int wave = threadIdx.x >> 5, lane = threadIdx.x & 31;
    const int n0 = wave * 16;
    v8f acc = wmma_tile_f32<HDIM>(As, W, HDIM, n0, lane);
    const int half = lane >> 4, mr = lane & 15;
#pragma unroll
    for (int r = 0; r < 8; ++r) {
        const int m = m0 + r + 8 * half;
        if (m < Nn) out[(size_t)m * HDIM + n0 + mr] = acc[r];
    }
}

// ---- Final GEMM: out = h @ W_lin + b_lin, K=128, ncol=16 -------------------
__global__ void k_gemm_out(const float* __restrict__ Hm,
                           const float* __restrict__ W,
                           const float* __restrict__ bias,
                           float* __restrict__ out, int Nn) {
    const int lane = threadIdx.x & 31;
    const int wg   = (blockIdx.x * blockDim.x + threadIdx.x) >> 5;
    const int m0   = wg * 16;
    if (m0 >= Nn) return;
    const int half = lane >> 4, mr = lane & 15;
    v8f acc = {0.f, 0.f, 0.f, 0.f, 0.f, 0.f, 0.f, 0.f};
#pragma unroll 8
    for (int k0 = 0; k0 < HDIM; k0 += 4) {
        const int ka  = k0 + 2 * half;
        const int row = min(m0 + mr, Nn - 1);
        v2f a, b;
        a.x = Hm[(size_t)row * HDIM + ka];
        a.y = Hm[(size_t)row * HDIM + ka + 1];
        b.x = W[(size_t)ka * 16 + mr];
        b.y = W[(size_t)(ka + 1) * 16 + mr];
        acc = __builtin_amdgcn_wmma_f32_16x16x4_f32(false, a, false, b,
                                                    (short)0, acc, false, false);
    }
    const float bv = bias[mr];
#pragma unroll
    for (int r = 0; r < 8; ++r) {
        const int m = m0 + r + 8 * half;
        if (m < Nn) out[(size_t)m * 16 + mr] = acc[r] + bv;
    }
}

// ---- Degree / norm ---------------------------------------------------------
__global__ void k_deg_init(float* __restrict__ deg, int Nn) {
    const int i = blockIdx.x * blockDim.x + threadIdx.x;
    if (i < Nn) deg[i] = 1.0f;  // self loop
}
__global__ void k_deg_acc(const int* __restrict__ col, float* __restrict__ deg, int E) {
    const int e = blockIdx.x * blockDim.x + threadIdx.x;
    if (e < E) atomicAdd(&deg[col[e]], 1.0f);
}
__global__ void k_dinv(float* __restrict__ deg, int Nn) {
    const int i = blockIdx.x * blockDim.x + threadIdx.x;
    if (i < Nn) deg[i] = rsqrtf(deg[i]);  // deg >= 1 always (self loop)
}

// ---- Conv: out = b + dinv[i]^2 * t[i]  (self-loop term + bias) -------------
__global__ void k_conv_init(const float* __restrict__ t,
                            const float* __restrict__ dinv,
                            const float* __restrict__ bias,
                            float* __restrict__ out, int Nn) {
    const size_t idx = (size_t)blockIdx.x * blockDim.x + threadIdx.x;
    if (idx >= (size_t)Nn * HDIM) return;
    const int i = (int)(idx >> 7), f = (int)(idx & (HDIM - 1));
    const float d = dinv[i];
    out[idx] = bias[f] + d * d * t[idx];
}

// ---- Edge scatter: out[col] += dinv[row]*dinv[col] * t[row] ----------------
// One wave32 per edge; each lane moves a float4 (128 f32 row total).
__global__ void k_edge_scatter(const int* __restrict__ row,
                               const int* __restrict__ col,
                               const float* __restrict__ t,
                               const float* __restrict__ dinv,
                               float* __restrict__ out, int E) {
    const int e    = (int)(((size_t)blockIdx.x * blockDim.x + threadIdx.x) >> 5);
    const int lane = threadIdx.x & 31;
    if (e >= E) return;
    const int r = row[e], c = col[e];
    const float w = dinv[r] * dinv[c];
    const float4 v = reinterpret_cast<const float4*>(t + (size_t)r * HDIM)[lane];
    float* dst = out + (size_t)c * HDIM + lane * 4;
    atomicAdd(dst + 0, v.x * w);
    atomicAdd(dst + 1, v.y * w);
    atomicAdd(dst + 2, v.z * w);
    atomicAdd(dst + 3, v.w * w);
}

// ---- BatchNorm -------------------------------------------------------------
__global__ void k_zero_stats(float* __restrict__ stats) {
    stats[threadIdx.x] = 0.0f;  // 256 threads: [0..127]=sum, [128..255]=sumsq
}
__global__ void k_bn_stats(const float* __restrict__ x,
                           float* __restrict__ stats, int Nn) {
    const int f = threadIdx.x;  // blockDim.x == 128
    float s = 0.0f, s2 = 0.0f;
    for (int i = blockIdx.x; i < Nn; i += gridDim.x) {
        const float v = x[(size_t)i * HDIM + f];
        s += v;
        s2 += v * v;
    }
    atomicAdd(&stats[f], s);
    atomicAdd(&stats[HDIM + f], s2);
}
__global__ void k_bn_selu_res(const float* __restrict__ x,
                              const float* __restrict__ stats,
                              const float* __restrict__ g,
                              const float* __restrict__ be,
                              const float* __restrict__ identity,
                              float* __restrict__ out, int Nn) {
    const size_t idx = (size_t)blockIdx.x * blockDim.x + threadIdx.x;
    if (idx >= (size_t)Nn * HDIM) return;
    const int f = (int)(idx & (HDIM - 1));
    const float invN = 1.0f / (float)Nn;
    const float mu  = stats[f] * invN;
    float var = stats[HDIM + f] * invN - mu * mu;
    var = var > 0.0f ? var : 0.0f;
    const float rstd = rsqrtf(var + 1e-5f);
    const float y = g[f] * (x[idx] - mu) * rstd + be[f];
    out[idx] = selu_f(y) + identity[idx];
}

// ---------------------------------------------------------------------------
extern "C" void kernel_launch(void* const* d_in, const int* in_sizes, int n_in,
                              void* d_out, int out_size, void* d_ws, size_t ws_size,
                              hipStream_t stream) {
    const float* x     = (const float*)d_in[0];
    const int*   eidx  = (const int*)d_in[1];
    const float* W_red = (const float*)d_in[2];
    const float* b_red = (const float*)d_in[3];
    const float* W1    = (const float*)d_in[4];
    const float* b1    = (const float*)d_in[5];
    const float* g1    = (const float*)d_in[6];
    const float* be1   = (const float*)d_in[7];
    const float* W2    = (const float*)d_in[8];
    const float* b2    = (const float*)d_in[9];
    const float* g2    = (const float*)d_in[10];
    const float* be2   = (const float*)d_in[11];
    const float* W_lin = (const float*)d_in[12];
    const float* b_lin = (const float*)d_in[13];

    const int N = in_sizes[0] / FIN;
    const int E = in_sizes[1] / 2;
    const int* row = eidx;
    const int* col = eidx + E;

    const size_t NH = (size_t)N * HDIM;
    float* h0    = (float*)d_ws;      // identity / layer buffers
    float* t     = h0 + NH;           // GEMM output (pre-propagation)
    float* cv    = t + NH;            // conv accumulator / layer output
    float* dinv  = cv + NH;           // N floats: degree -> rsqrt(degree)
    float* stats = dinv + N;          // 256 floats: BN sums

    const int tilesM   = (N + 15) / 16;
    const int ewBlocks = (E + 7) / 8;       // 8 waves (edges) per 256-thread block
    const int nhBlocks = (int)((NH + 255) / 256);
    const int nBlocks  = (N + 255) / 256;
    const int eBlocks  = (E + 255) / 256;

    // Layer 0: h0 = selu(x @ W_red + b_red)
    k_gemm_selu<<<tilesM, 256, 0, stream>>>(x, W_red, b_red, h0, N);

    // Normalization coefficients (shared by both conv layers)
    k_deg_init<<<nBlocks, 256, 0, stream>>>(dinv, N);
    k_deg_acc<<<eBlocks, 256, 0, stream>>>(col, dinv, E);
    k_dinv<<<nBlocks, 256, 0, stream>>>(dinv, N);

    // ---- Layer 1: cv = selu(bn(gcn(h0, W1))) + h0 ----
    k_gemm_h<<<tilesM, 256, 0, stream>>>(h0, W1, t, N);
    k_conv_init<<<nhBlocks, 256, 0, stream>>>(t, dinv, b1, cv, N);
    k_edge_scatter<<<ewBlocks, 256, 0, stream>>>(row, col, t, dinv, cv, E);
    k_zero_stats<<<1, 256, 0, stream>>>(stats);
    k_bn_stats<<<1024, 128, 0, stream>>>(cv, stats, N);
    k_bn_selu_res<<<nhBlocks, 256, 0, stream>>>(cv, stats, g1, be1, h0, cv, N);

    // ---- Layer 2: h0 = selu(bn(gcn(cv, W2))) + cv ----
    k_gemm_h<<<tilesM, 256, 0, stream>>>(cv, W2, t, N);
    k_conv_init<<<nhBlocks, 256, 0, stream>>>(t, dinv, b2, h0, N);
    k_edge_scatter<<<ewBlocks, 256, 0, stream>>>(row, col, t, dinv, h0, E);
    k_zero_stats<<<1, 256, 0, stream>>>(stats);
    k_bn_stats<<<1024, 128, 0, stream>>>(h0, stats, N);
    k_bn_selu_res<<<nhBlocks, 256, 0, stream>>>(h0, stats, g2, be2, cv, h0, N);

    // Final linear: d_out = h0 @ W_lin + b_lin
    const int outBlocks = (tilesM + 7) / 8;
    k_gemm_out<<<outBlocks, 256, 0, stream>>>(h0, W_lin, b_lin, (float*)d_out, N);
}